// HMMAligner_56392920596482
// MI455X (gfx1250) — compile-verified
//
#include <hip/hip_runtime.h>
#include <hip/hip_bf16.h>
#include <stdint.h>

// HMM forward (forced-alignment) for B=64, T=1000, V=512, U=256.
// One wave32 workgroup per batch element; 8 states/lane in registers.
// Emissions streamed HBM -> LDS via CDNA5 Tensor Data Mover, 8-deep ring
// (128 KB LDS) to hide DMA latency; synchronized with s_wait_tensorcnt.

#define B_DIM 64
#define T_DIM 1000
#define V_DIM 512
#define U_DIM 256
#define CH    8              // time rows per TDM tile (1000 % 8 == 0)
#define RING  8              // ring depth: 8 x 16KB = 128KB LDS, 7 chunks in flight
#define SPL   8              // states per lane (8 * 32 = 256 = U)
#define NCH   (T_DIM / CH)   // 125 chunks

#define NEG_INF   (-1.0e5f)
#define LOG_HALF  (-0.69314718055994530942f)

typedef uint32_t v4u __attribute__((ext_vector_type(4)));
typedef int      v8i __attribute__((ext_vector_type(8)));
typedef int      v4i __attribute__((ext_vector_type(4)));

// Issue one TDM load: tile of CH rows x V_DIM f32 from log_posteriors[b, t0.., :]
// into an LDS ring slot. D# packing per CDNA5 ISA.
__device__ __forceinline__ void tdm_issue(const float* lp, int b, int t0,
                                          float* lds_dst) {
  uint64_t ga = (uint64_t)(uintptr_t)(lp + ((size_t)b * T_DIM + t0) * V_DIM);
  uint32_t ga_lo = (uint32_t)ga;
  uint32_t ga_hi = (uint32_t)(ga >> 32);
  // Flat shared-space pointers carry the LDS byte offset in addr[31:0].
  uint32_t lds_addr = (uint32_t)(uintptr_t)lds_dst;

  // ---- D# group 0 (128b): count=1 | lds_addr | global_addr[56:0] | type=2 ----
  v4u g0;
  g0.x = 1u;                                          // count=1, user mode
  g0.y = lds_addr;                                    // bits 63:32
  g0.z = ga_lo;                                       // bits 95:64
  g0.w = (ga_hi & 0x01FFFFFFu) | (2u << 30);          // addr[56:32] | type=2

  // ---- D# group 1 (256b) ----
  int rows = CH;
  if (t0 + rows > T_DIM) rows = T_DIM - t0;
  v8i g1;
  g1[0] = 0x00020000;                 // workgroup_mask=0, data_size=2 (4B), no pad/iter/barrier
  g1[1] = (int)((V_DIM & 0xFFFF) << 16);          // [63:48]  tensor_dim0 lo16 = 512
  g1[2] = (int)((V_DIM >> 16) |                   // [79:64]  tensor_dim0 hi16 = 0
                ((T_DIM & 0xFFFF) << 16));        // [95:80]  tensor_dim1 lo16 = 1000
  g1[3] = (int)((T_DIM >> 16) |                   // [111:96] tensor_dim1 hi16 = 0
                ((V_DIM & 0xFFFF) << 16));        // [127:112] tile_dim0 = 512
  g1[4] = (int)(rows & 0xFFFF);                   // [143:128] tile_dim1 = rows; tile_dim2 = 0
  g1[5] = V_DIM;                                  // [191:160] tensor_dim0_stride lo32 = 512
  g1[6] = 0;                                      // stride hi16; tensor_dim1_stride lo16 = 0
  g1[7] = 0;                                      // tensor_dim1_stride hi32 = 0

  v4i gz = {0, 0, 0, 0};                          // groups 2/3: unused (<=2D tensor)
#if defined(__clang_major__) && (__clang_major__ >= 23)
  v8i gz8 = {0, 0, 0, 0, 0, 0, 0, 0};
  __builtin_amdgcn_tensor_load_to_lds(g0, g1, gz, gz, gz8, 0);
#else
  __builtin_amdgcn_tensor_load_to_lds(g0, g1, gz, gz, 0);
#endif
}

__global__ __launch_bounds__(32) void hmm_forward_kernel(
    const float* __restrict__ lp,        // (B,T,V)
    const float* __restrict__ lens,      // (B,)
    const int*   __restrict__ phns,      // (B,U)
    const float* __restrict__ phn_lens,  // (B,)
    float* __restrict__ out)             // (B,)
{
  __shared__ float sm_e[RING * CH * V_DIM];   // 128 KB emission-row ring buffer

  const int b    = blockIdx.x;
  const int lane = threadIdx.x;

  const int Tval = (int)roundf(lens[b] * (float)T_DIM);
  const int Uval = (int)roundf(phn_lens[b] * (float)U_DIM);

  // Phoneme ids for this lane's 8 states (kept in registers).
  int ph[SPL];
#pragma unroll
  for (int k = 0; k < SPL; ++k)
    ph[k] = phns[(size_t)b * U_DIM + lane * SPL + k];

  float a[SPL];        // alpha for states u = lane*SPL + k

  // ---- Prologue: fill the DMA ring (chunks 0..RING-1 in flight). ----
#pragma unroll
  for (int i = 0; i < RING; ++i)
    tdm_issue(lp, b, i * CH, &sm_e[i * CH * V_DIM]);

  // One chunk's worth of DP steps, reading emissions from an LDS ring slot.
  auto process_chunk = [&](int c) {
    const float* ebase = &sm_e[(c % RING) * CH * V_DIM];
#pragma unroll
    for (int r = 0; r < CH; ++r) {
      const int t = c * CH + r;
      const float* row = ebase + r * V_DIM;

      // Gather masked emissions e[u] = row[phns[u]] (ds_load_b32 gather).
      float e[SPL];
#pragma unroll
      for (int k = 0; k < SPL; ++k) {
        int u   = lane * SPL + k;
        float v = row[ph[k]];
        e[k] = (u < Uval && t < Tval) ? v : NEG_INF;
      }

      if (t == 0) {
        // alpha0 = pi + e0 (pi: 0 at u==0, else NEG_INF; unclamped like reference)
#pragma unroll
        for (int k = 0; k < SPL; ++k) {
          int u = lane * SPL + k;
          a[k] = ((u == 0) ? 0.0f : NEG_INF) + e[k];
        }
      } else {
        // Banded log-matvec: x_u = LSE(a_{u-1}+log1/2, a_u+(u==Uval-1?0:log1/2)).
        float boundary = __shfl_up(a[SPL - 1], 1, 32);
        float prev_in  = (lane == 0) ? -INFINITY : boundary;
        float na[SPL];
#pragma unroll
        for (int k = 0; k < SPL; ++k) {
          int u      = lane * SPL + k;
          float am1  = (k == 0) ? prev_in : a[k - 1];
          float self = a[k] + ((u == Uval - 1) ? 0.0f : LOG_HALF);
          float prev = am1 + LOG_HALF;
          float mx = fmaxf(self, prev);
          float mn = fminf(self, prev);
          float x  = mx + __logf(1.0f + __expf(mn - mx));  // v_exp_f32 / v_log_f32
          na[k] = (u < Uval) ? (x + e[k]) : NEG_INF;       // where(m, x+e, NEG_INF)
        }
#pragma unroll
        for (int k = 0; k < SPL; ++k) a[k] = na[k];
      }
    }
  };

  // ---- Main loop: keep RING-1 chunks in flight. ----
  for (int c = 0; c < NCH - RING; ++c) {
    __builtin_amdgcn_s_wait_tensorcnt(RING - 1);   // chunk c resident (in-order)
    asm volatile("" ::: "memory");                 // TDM writes invisible to AA
    process_chunk(c);
    // Ring slot (c+RING)%RING == c%RING is now free; make sure all ds reads
    // of it retired before the DMA engine can overwrite it.
    asm volatile("s_wait_dscnt 0" ::: "memory");
    tdm_issue(lp, b, (c + RING) * CH, &sm_e[(c % RING) * CH * V_DIM]);
  }

  // ---- Tail: everything already issued; drain and finish. ----
  __builtin_amdgcn_s_wait_tensorcnt(0);
  asm volatile("" ::: "memory");
  for (int c = NCH - RING; c < NCH; ++c)
    process_chunk(c);

  // ---- Final logsumexp over all 256 states. ----
  float m = a[0];
#pragma unroll
  for (int k = 1; k < SPL; ++k) m = fmaxf(m, a[k]);
  for (int off = 16; off > 0; off >>= 1) m = fmaxf(m, __shfl_xor(m, off, 32));

  float s = 0.0f;
#pragma unroll
  for (int k = 0; k < SPL; ++k) s += __expf(a[k] - m);
  for (int off = 16; off > 0; off >>= 1) s += __shfl_xor(s, off, 32);

  if (lane == 0) out[b] = m + __logf(s);
}

extern "C" void kernel_launch(void* const* d_in, const int* in_sizes, int n_in,
                              void* d_out, int out_size, void* d_ws, size_t ws_size,
                              hipStream_t stream) {
  const float* lp       = (const float*)d_in[0];  // log_posteriors (B,T,V) f32
  const float* lens     = (const float*)d_in[1];  // (B,) f32
  const int*   phns     = (const int*)d_in[2];    // (B,U) i32
  const float* phn_lens = (const float*)d_in[3];  // (B,) f32
  float*       out      = (float*)d_out;          // (B,) f32

  const int B = in_sizes[1];                      // 64
  hmm_forward_kernel<<<B, 32, 0, stream>>>(lp, lens, phns, phn_lens, out);
}